// SpikeFP32SiLU_23407571764130
// MI455X (gfx1250) — compile-verified
//
#include <hip/hip_runtime.h>

// SiLU(x) = x * sigmoid(x), elementwise over 134M fp32 values.
// HBM-streaming workload: 1.07 GB traffic -> ~46us floor at 23.3 TB/s.
// Strategy: B128 non-temporal loads/stores (single-touch data, don't thrash
// the 192MB L2), 4 independent in-flight B128 loads per wave32 thread, and
// hardware v_exp_f32 / v_rcp_f32 for the sigmoid.
//
// Addressing: buffer is 536MB < 2^31 B, so we use 32-bit element indices with
// FOUR uniform base pointers (in + k*stride), hoisted out of the loop. Each
// access is then gep(uniform_base, zext(i)) which the gfx1250 backend lowers
// to the GVS saddr form: global_load_b128 vdst, v_idx, s[base] scale_offset.
// The loop body carries a single 32-bit index VGPR instead of per-lane 64-bit
// pointer chains.

typedef float f4 __attribute__((ext_vector_type(4)));

__device__ __forceinline__ float silu_fast(float x) {
    // sigmoid(x) = 1 / (1 + 2^(-x * log2(e)))
    // v_exp_f32 is base-2 on AMD; limits behave correctly:
    //   x -> -inf : exp2 -> +inf, rcp -> 0, result -> 0
    //   x -> +inf : exp2 -> 0,    rcp(1) -> 1, result -> x
    const float NLOG2E = -1.4426950408889634f;
    float e = __builtin_amdgcn_exp2f(x * NLOG2E);
    return x * __builtin_amdgcn_rcpf(1.0f + e);
}

__device__ __forceinline__ f4 silu4(f4 v) {
    f4 r;
    r.x = silu_fast(v.x);
    r.y = silu_fast(v.y);
    r.z = silu_fast(v.z);
    r.w = silu_fast(v.w);
    return r;
}

__global__ __launch_bounds__(256) void silu_v4_kernel(const f4* __restrict__ in,
                                                      f4* __restrict__ out,
                                                      unsigned nvec) {
    const unsigned stride = gridDim.x * blockDim.x;      // uniform (SGPR)
    unsigned i = blockIdx.x * blockDim.x + threadIdx.x;  // 32-bit lane index

    // Uniform base pointers: computed once on SALU, enabling saddr-form
    // (SGPR64 base + shared 32-bit VGPR index, scale_offset) for every
    // load/store in the unrolled body.
    const f4* __restrict__ in0 = in;
    const f4* __restrict__ in1 = in + stride;
    const f4* __restrict__ in2 = in + 2u * stride;
    const f4* __restrict__ in3 = in + 3u * stride;
    f4* __restrict__ out0 = out;
    f4* __restrict__ out1 = out + stride;
    f4* __restrict__ out2 = out + 2u * stride;
    f4* __restrict__ out3 = out + 3u * stride;

    // 4-wide unrolled grid-stride loop: 4 independent NT B128 loads issued
    // back-to-back (LOADcnt -> 4) before the first dependent use, keeping
    // 4 cachelines per wave in flight toward HBM.
    for (; i + 3u * stride < nvec; i += 4u * stride) {
        f4 a = __builtin_nontemporal_load(in0 + i);
        f4 b = __builtin_nontemporal_load(in1 + i);
        f4 c = __builtin_nontemporal_load(in2 + i);
        f4 d = __builtin_nontemporal_load(in3 + i);
        __builtin_nontemporal_store(silu4(a), out0 + i);
        __builtin_nontemporal_store(silu4(b), out1 + i);
        __builtin_nontemporal_store(silu4(c), out2 + i);
        __builtin_nontemporal_store(silu4(d), out3 + i);
    }
    // Remainder vectors (nvec not a multiple of 4*stride).
    for (; i < nvec; i += stride) {
        f4 a = __builtin_nontemporal_load(in0 + i);
        __builtin_nontemporal_store(silu4(a), out0 + i);
    }
}

// Scalar tail for n % 4 elements (zero for this shape; kept for generality).
__global__ __launch_bounds__(64) void silu_tail_kernel(const float* __restrict__ in,
                                                       float* __restrict__ out,
                                                       int n) {
    int i = blockIdx.x * blockDim.x + threadIdx.x;
    if (i < n) {
        out[i] = silu_fast(in[i]);
    }
}

extern "C" void kernel_launch(void* const* d_in, const int* in_sizes, int n_in,
                              void* d_out, int out_size, void* d_ws, size_t ws_size,
                              hipStream_t stream) {
    (void)n_in; (void)d_ws; (void)ws_size; (void)out_size;

    const float* x = (const float*)d_in[0];
    float* out = (float*)d_out;

    const unsigned n    = (unsigned)in_sizes[0];   // 8*8192*2048 = 134,217,728
    const unsigned nvec = n >> 2;                  // float4 count (33,554,432)
    const unsigned tail = n - (nvec << 2);

    if (nvec > 0) {
        const int block = 256;                     // 8 waves (wave32) per block
        unsigned want = (nvec + block - 1) / block;
        int grid = (int)(want < 4096u ? want : 4096u); // 1M threads, 32 f4/thread
        if (grid < 1) grid = 1;
        silu_v4_kernel<<<grid, block, 0, stream>>>((const f4*)x, (f4*)out, nvec);
    }
    if (tail > 0) {
        silu_tail_kernel<<<1, 64, 0, stream>>>(x + ((size_t)nvec << 2),
                                               out + ((size_t)nvec << 2), (int)tail);
    }
}